// AdditiveAttention_60790967107835
// MI455X (gfx1250) — compile-verified
//
#include <hip/hip_runtime.h>
#include <hip/hip_bf16.h>

typedef __attribute__((ext_vector_type(2))) float v2f;
typedef __attribute__((ext_vector_type(8))) float v8f;

#define B_DIM 128
#define T_DIM 4096
#define D_K 16
#define D_W 32
#define NTHREADS 256
#define NWAVES 8
#define NTILES (T_DIM / 16)   // 256 tiles of 16 timesteps

// Hardware tanh (V_TANH_F32 on gfx1250) with libm fallback.
__device__ __forceinline__ float fast_tanh(float x) {
#if __has_builtin(__builtin_amdgcn_tanhf)
    return __builtin_amdgcn_tanhf(x);
#elif __has_builtin(__builtin_amdgcn_tanh_f32)
    return __builtin_amdgcn_tanh_f32(x);
#else
    return tanhf(x);
#endif
}

// Fast exp for x <= 0 (softmax: score - max): single V_EXP_F32 (2^x).
__device__ __forceinline__ float fast_exp_nonpos(float x) {
#if __has_builtin(__builtin_amdgcn_exp2f)
    return __builtin_amdgcn_exp2f(x * 1.4426950408889634f);
#else
    return __expf(x);
#endif
}

// Cross-lane xor moves. Prefer VALU DPP (no LDS round-trip / dscnt wait) over
// ds_bpermute. xor1/2/4 stay within 8-lane groups -> DPP8; xor8 = row_ror:8
// within the 16-lane DPP row (rotate by 8 of 16 == xor 8).
#if __has_builtin(__builtin_amdgcn_mov_dpp8) && __has_builtin(__builtin_amdgcn_update_dpp)
#define DPP8_SEL(a,b,c,d,e,f,g,h) \
    ((a)|((b)<<3)|((c)<<6)|((d)<<9)|((e)<<12)|((f)<<15)|((g)<<18)|((h)<<21))
__device__ __forceinline__ float lane_xor1(float x) {
    return __int_as_float(__builtin_amdgcn_mov_dpp8(__float_as_int(x),
        DPP8_SEL(1,0,3,2,5,4,7,6)));
}
__device__ __forceinline__ float lane_xor2(float x) {
    return __int_as_float(__builtin_amdgcn_mov_dpp8(__float_as_int(x),
        DPP8_SEL(2,3,0,1,6,7,4,5)));
}
__device__ __forceinline__ float lane_xor4(float x) {
    return __int_as_float(__builtin_amdgcn_mov_dpp8(__float_as_int(x),
        DPP8_SEL(4,5,6,7,0,1,2,3)));
}
__device__ __forceinline__ float lane_xor8(float x) {
    int i = __float_as_int(x);
    return __int_as_float(__builtin_amdgcn_update_dpp(i, i, 0x128 /*row_ror:8*/,
                                                      0xf, 0xf, false));
}
#else
__device__ __forceinline__ float lane_xor1(float x) { return __shfl_xor(x, 1); }
__device__ __forceinline__ float lane_xor2(float x) { return __shfl_xor(x, 2); }
__device__ __forceinline__ float lane_xor4(float x) { return __shfl_xor(x, 4); }
__device__ __forceinline__ float lane_xor8(float x) { return __shfl_xor(x, 8); }
#endif

// One block per batch. 8 waves; each wave computes scores for 32 tiles of 16
// timesteps using V_WMMA_F32_16X16X4_F32 (K=16 via 4 chained K=4 WMMAs,
// D_W=32 via two N=16 tiles). Scores -> LDS, block softmax, then stream value
// for the context reduction.
__global__ __launch_bounds__(NTHREADS) void AdditiveAttention_60790967107835_kernel(
    const float* __restrict__ query,   // [B,1,16]
    const float* __restrict__ key,     // [B,T,16]
    const float* __restrict__ value,   // [B,T,16]
    const float* __restrict__ W1,      // [32,16]
    const float* __restrict__ W2,      // [32,16]
    const float* __restrict__ bias,    // scalar
    const float* __restrict__ v_w,     // [1,32]
    const float* __restrict__ v_b,     // scalar
    float* __restrict__ out)           // [B*16] context ++ [B*T] attn
{
    __shared__ float s_scores[T_DIM];   // 16 KB
    __shared__ float s_red[NTHREADS];
    __shared__ float s_ctx[D_K];

    const int b    = blockIdx.x;
    const int tid  = threadIdx.x;
    const int lane = tid & 31;
    const int wave = tid >> 5;

    const int col = lane & 15;          // N (or M row for A) index within tile
    const int kb  = (lane >> 4) << 1;   // 0 for lanes 0-15, 2 for lanes 16-31

    // ---- B-matrix fragments: B[K][N] = W2[wofs+N][K], K-chunked by 4 ----
    v2f bfrag[2][4];
#pragma unroll
    for (int j = 0; j < 2; ++j) {
        const float* wrow = W2 + (size_t)(16 * j + col) * D_K;
#pragma unroll
        for (int kc = 0; kc < 4; ++kc)
            bfrag[j][kc] = *(const v2f*)(wrow + 4 * kc + kb);
    }

    // ---- q_proj[w] = sum_d query[b,d] * W1[w,d], for w = col and col+16 ----
    const float* qb = query + (size_t)b * D_K;
    float qp[2];
#pragma unroll
    for (int j = 0; j < 2; ++j) {
        const float* w1row = W1 + (size_t)(16 * j + col) * D_K;
        float s = 0.0f;
#pragma unroll
        for (int d = 0; d < D_K; ++d) s = fmaf(qb[d], w1row[d], s);
        qp[j] = s;
    }

    const float vw0 = v_w[col];
    const float vw1 = v_w[16 + col];
    const float biasv = bias[0];
    const float vbv = v_b[0];

    const float add0 = qp[0] + biasv;
    const float add1 = qp[1] + biasv;

    // ---- scores via WMMA: tile = 16 timesteps, wave-strided over 256 tiles ----
    const float* keyb = key + (size_t)b * T_DIM * D_K;
    for (int tile = wave; tile < NTILES; tile += NWAVES) {
        const int t0 = tile * 16;
        const float* kp = keyb + (size_t)(t0 + col) * D_K;  // A row = col

        // speculative prefetch of this wave's next tile (near-cache hint)
        __builtin_prefetch(keyb + (size_t)(((tile + NWAVES) & (NTILES - 1)) * 16 + col) * D_K, 0, 3);

        v8f c0 = {};
        v8f c1 = {};
#pragma unroll
        for (int kc = 0; kc < 4; ++kc) {
            v2f a = *(const v2f*)(kp + 4 * kc + kb);
            c0 = __builtin_amdgcn_wmma_f32_16x16x4_f32(
                false, a, false, bfrag[0][kc], (short)0, c0, false, false);
            c1 = __builtin_amdgcn_wmma_f32_16x16x4_f32(
                false, a, false, bfrag[1][kc], (short)0, c1, false, false);
        }

        // h = tanh(k_proj + q_proj + bias); per-lane score contribution
        float p[8];
#pragma unroll
        for (int r = 0; r < 8; ++r) {
            float h0 = fast_tanh(c0[r] + add0);
            float h1 = fast_tanh(c1[r] + add1);
            p[r] = fmaf(h0, vw0, h1 * vw1);
        }

        // Tree reduction: fold the 16-lane N dimension while packing rows
        // into lanes. After the tree, lane n holds the full score for row
        // (n&7) (lanes 0-15) or 8+(n&7) (lanes 16-31).
        float q4[4];
#pragma unroll
        for (int i = 0; i < 4; ++i) {
            float a = p[2 * i]     + lane_xor1(p[2 * i]);
            float c = p[2 * i + 1] + lane_xor1(p[2 * i + 1]);
            q4[i] = (lane & 1) ? c : a;
        }
        float q2[2];
#pragma unroll
        for (int i = 0; i < 2; ++i) {
            float a = q4[2 * i]     + lane_xor2(q4[2 * i]);
            float c = q4[2 * i + 1] + lane_xor2(q4[2 * i + 1]);
            q2[i] = (lane & 2) ? c : a;
        }
        {
            float a = q2[0] + lane_xor4(q2[0]);
            float c = q2[1] + lane_xor4(q2[1]);
            float z = (lane & 4) ? c : a;
            z += lane_xor8(z);
            // one coalesced predicated store: lanes 0-7 -> rows 0-7,
            // lanes 16-23 -> rows 8-15
            if ((lane & 8) == 0)
                s_scores[t0 + (lane & 7) + ((lane & 16) >> 1)] = z + vbv;
        }
    }
    __syncthreads();

    // ---- softmax: block max ----
    float m = -INFINITY;
    for (int t = tid; t < T_DIM; t += NTHREADS) m = fmaxf(m, s_scores[t]);
    s_red[tid] = m;
    __syncthreads();
    for (int s = NTHREADS / 2; s > 0; s >>= 1) {
        if (tid < s) s_red[tid] = fmaxf(s_red[tid], s_red[tid + s]);
        __syncthreads();
    }
    const float gmax = s_red[0];
    __syncthreads();

    // ---- exp and block sum ----
    float lsum = 0.0f;
    for (int t = tid; t < T_DIM; t += NTHREADS) {
        float e = fast_exp_nonpos(s_scores[t] - gmax);
        s_scores[t] = e;
        lsum += e;
    }
    s_red[tid] = lsum;
    if (tid < D_K) s_ctx[tid] = 0.0f;
    __syncthreads();
    for (int s = NTHREADS / 2; s > 0; s >>= 1) {
        if (tid < s) s_red[tid] += s_red[tid + s];
        __syncthreads();
    }
    const float inv = 1.0f / s_red[0];

    // ---- attn writeback + context accumulation ----
    float ctx[D_K];
#pragma unroll
    for (int d = 0; d < D_K; ++d) ctx[d] = 0.0f;

    float* attn_out = out + (size_t)B_DIM * D_K + (size_t)b * T_DIM;
    const float* valb = value + (size_t)b * T_DIM * D_K;
    for (int t = tid; t < T_DIM; t += NTHREADS) {
        float a = s_scores[t] * inv;
        attn_out[t] = a;
        const float4* vp = (const float4*)(valb + (size_t)t * D_K);
#pragma unroll
        for (int i = 0; i < 4; ++i) {
            float4 v = vp[i];
            ctx[4 * i + 0] = fmaf(a, v.x, ctx[4 * i + 0]);
            ctx[4 * i + 1] = fmaf(a, v.y, ctx[4 * i + 1]);
            ctx[4 * i + 2] = fmaf(a, v.z, ctx[4 * i + 2]);
            ctx[4 * i + 3] = fmaf(a, v.w, ctx[4 * i + 3]);
        }
    }
#pragma unroll
    for (int d = 0; d < D_K; ++d) atomicAdd(&s_ctx[d], ctx[d]);
    __syncthreads();
    if (tid < D_K) out[(size_t)b * D_K + tid] = s_ctx[tid];
}

extern "C" void kernel_launch(void* const* d_in, const int* in_sizes, int n_in,
                              void* d_out, int out_size, void* d_ws, size_t ws_size,
                              hipStream_t stream) {
    const float* query = (const float*)d_in[0];
    const float* key   = (const float*)d_in[1];
    const float* value = (const float*)d_in[2];
    const float* W1    = (const float*)d_in[3];
    const float* W2    = (const float*)d_in[4];
    const float* bias  = (const float*)d_in[5];
    const float* v_w   = (const float*)d_in[6];
    const float* v_b   = (const float*)d_in[7];
    float* out = (float*)d_out;

    dim3 grid(B_DIM);
    dim3 block(NTHREADS);
    AdditiveAttention_60790967107835_kernel<<<grid, block, 0, stream>>>(
        query, key, value, W1, W2, bias, v_w, v_b, out);
}